// GNN_77335181132165
// MI455X (gfx1250) — compile-verified
//
#include <hip/hip_runtime.h>
#include <math.h>

typedef __attribute__((ext_vector_type(2))) float v2f;
typedef __attribute__((ext_vector_type(8))) float v8f;

#define DH 32

// ---------------------------------------------------------------- utilities
__global__ void zero_kernel(float* __restrict__ p, long n) {
  long t = (long)blockIdx.x * blockDim.x + threadIdx.x;
  if (t < n) p[t] = 0.0f;
}

// ------------------------------------------------- layer-1 fused transform
// Tl = X @ Wl            (message transform, no bias)
// Tr = X @ Wr + bias_r   (self transform)
// X: [nrows,64] row-major, W: [64,32], outputs [nrows,32]. nrows % 16 == 0.
// One wave -> 16 rows x 32 cols (two 16x16 tiles per output).
__global__ void gemm64_dual_wmma(const float* __restrict__ X,
                                 const float* __restrict__ Wl,
                                 const float* __restrict__ Wr,
                                 const float* __restrict__ bias_r,
                                 float* __restrict__ Tl,
                                 float* __restrict__ Tr,
                                 int nrows) {
  const int K = 64;
  int wave = blockIdx.x * (blockDim.x >> 5) + (threadIdx.x >> 5);
  int lane = threadIdx.x & 31;
  int row0 = wave << 4;
  if (row0 >= nrows) return;          // wave-uniform; EXEC stays all-ones
  int half = lane >> 4;               // 0: K offset 0,1 ; 1: K offset 2,3
  int m    = lane & 15;
  const float* xrow = X + (long)(row0 + m) * K;

  v8f al0 = {}, al1 = {};
  v8f ar0, ar1;
  float b0 = bias_r[m], b1 = bias_r[m + 16];
#pragma unroll
  for (int i = 0; i < 8; ++i) { ar0[i] = b0; ar1[i] = b1; }

#pragma unroll
  for (int k = 0; k < K; k += 4) {
    int kk = k + half * 2;
    v2f a;  a.x = xrow[kk];  a.y = xrow[kk + 1];
    v2f bl0, bl1, br0, br1;
    bl0.x = Wl[kk * DH + m];        bl0.y = Wl[(kk + 1) * DH + m];
    bl1.x = Wl[kk * DH + m + 16];   bl1.y = Wl[(kk + 1) * DH + m + 16];
    br0.x = Wr[kk * DH + m];        br0.y = Wr[(kk + 1) * DH + m];
    br1.x = Wr[kk * DH + m + 16];   br1.y = Wr[(kk + 1) * DH + m + 16];
    al0 = __builtin_amdgcn_wmma_f32_16x16x4_f32(false, a, false, bl0, (short)0, al0, false, false);
    al1 = __builtin_amdgcn_wmma_f32_16x16x4_f32(false, a, false, bl1, (short)0, al1, false, false);
    ar0 = __builtin_amdgcn_wmma_f32_16x16x4_f32(false, a, false, br0, (short)0, ar0, false, false);
    ar1 = __builtin_amdgcn_wmma_f32_16x16x4_f32(false, a, false, br1, (short)0, ar1, false, false);
  }
#pragma unroll
  for (int r = 0; r < 8; ++r) {
    long rr = row0 + r + half * 8;   // C/D layout: vgpr r -> rows r / r+8
    Tl[rr * DH + m]      = al0[r];
    Tl[rr * DH + m + 16] = al1[r];
    Tr[rr * DH + m]      = ar0[r];
    Tr[rr * DH + m + 16] = ar1[r];
  }
}

// ------------------------------------------------- layer-2 transform (K=32)
// T = X @ W (+ bias if hasBias). X:[nrows,32], W:[32,32]. nrows % 16 == 0.
__global__ void gemm32_wmma(const float* __restrict__ X,
                            const float* __restrict__ W,
                            const float* __restrict__ bias,
                            float* __restrict__ T,
                            int nrows, int hasBias) {
  const int K = 32;
  int wave = blockIdx.x * (blockDim.x >> 5) + (threadIdx.x >> 5);
  int lane = threadIdx.x & 31;
  int row0 = wave << 4;
  if (row0 >= nrows) return;
  int half = lane >> 4;
  int m    = lane & 15;
  const float* xrow = X + (long)(row0 + m) * K;

  v8f a0, a1;
  float b0 = hasBias ? bias[m] : 0.0f;
  float b1 = hasBias ? bias[m + 16] : 0.0f;
#pragma unroll
  for (int i = 0; i < 8; ++i) { a0[i] = b0; a1[i] = b1; }

#pragma unroll
  for (int k = 0; k < K; k += 4) {
    int kk = k + half * 2;
    v2f a;  a.x = xrow[kk];  a.y = xrow[kk + 1];
    v2f w0, w1;
    w0.x = W[kk * DH + m];       w0.y = W[(kk + 1) * DH + m];
    w1.x = W[kk * DH + m + 16];  w1.y = W[(kk + 1) * DH + m + 16];
    a0 = __builtin_amdgcn_wmma_f32_16x16x4_f32(false, a, false, w0, (short)0, a0, false, false);
    a1 = __builtin_amdgcn_wmma_f32_16x16x4_f32(false, a, false, w1, (short)0, a1, false, false);
  }
#pragma unroll
  for (int r = 0; r < 8; ++r) {
    long rr = row0 + r + half * 8;
    T[rr * DH + m]      = a0[r];
    T[rr * DH + m + 16] = a1[r];
  }
}

// --------------------------------------------------------- edge scatter-add
// 8 lanes per edge, float4 per lane: one global_load_b128 gather + 4 dword
// f32 atomics per lane (no packed f32 atomic exists -> 32 atomics/edge is the
// hardware floor). 4 edges per wave. Feature tables (12.8 MB each) are
// L2-resident on the 192 MB L2, so both gather and atomics stay on-chip.
__global__ void scatter_edges(const int* __restrict__ edge, int E_,
                              const float* __restrict__ T,
                              float* __restrict__ agg,
                              float* __restrict__ cnt,
                              int addCount) {
  long t = (long)blockIdx.x * blockDim.x + threadIdx.x;
  long e = t >> 3;                       // 8 lanes per edge
  if (e >= E_) return;
  int j4  = (int)(t & 7) * 4;            // feature sub-offset 0,4,...,28
  int src = edge[e];
  int dst = edge[(long)E_ + e];
  float4 v = *(const float4*)(T + (long)src * DH + j4);   // 16B aligned
  float* ap = agg + (long)dst * DH + j4;
  __hip_atomic_fetch_add(ap + 0, v.x, __ATOMIC_RELAXED, __HIP_MEMORY_SCOPE_AGENT);
  __hip_atomic_fetch_add(ap + 1, v.y, __ATOMIC_RELAXED, __HIP_MEMORY_SCOPE_AGENT);
  __hip_atomic_fetch_add(ap + 2, v.z, __ATOMIC_RELAXED, __HIP_MEMORY_SCOPE_AGENT);
  __hip_atomic_fetch_add(ap + 3, v.w, __ATOMIC_RELAXED, __HIP_MEMORY_SCOPE_AGENT);
  if (addCount && (t & 7) == 0)
    __hip_atomic_fetch_add(&cnt[dst], 1.0f,
                           __ATOMIC_RELAXED, __HIP_MEMORY_SCOPE_AGENT);
}

// -------------------------------------------------- mean + self (+ relu)
// in place on agg: agg = relu( agg / max(cnt,1) + selfb )
__global__ void finalize_kernel(float* __restrict__ agg,
                                const float* __restrict__ selfb,
                                const float* __restrict__ cnt,
                                long n, int doRelu) {
  long t = (long)blockIdx.x * blockDim.x + threadIdx.x;
  if (t >= n) return;
  long i = t >> 5;
  float c = fmaxf(cnt[i], 1.0f);
  float v = agg[t] / c + selfb[t];
  if (doRelu) v = fmaxf(v, 0.0f);
  agg[t] = v;
}

// --------------------------------------------------------------- output head
// out[i] = sigmoid( dot( agg2[i]/max(cnt,1) + self2[i], Wlin ) + blin )
__global__ void head_kernel(const float* __restrict__ agg2,
                            const float* __restrict__ self2,
                            const float* __restrict__ cnt,
                            const float* __restrict__ Wlin,
                            const float* __restrict__ blin,
                            float* __restrict__ out, int nrows) {
  long t = (long)blockIdx.x * blockDim.x + threadIdx.x;
  long i = t >> 5;
  if (i >= nrows) return;
  int j = (int)(t & 31);
  float c = fmaxf(cnt[i], 1.0f);
  float v = (agg2[t] / c + self2[t]) * Wlin[j];
#pragma unroll
  for (int off = 16; off > 0; off >>= 1)
    v += __shfl_xor(v, off, 32);        // wave32 reduction
  if (j == 0)
    out[i] = 1.0f / (1.0f + expf(-(v + blin[0])));
}

// ============================================================= launch logic
extern "C" void kernel_launch(void* const* d_in, const int* in_sizes, int n_in,
                              void* d_out, int out_size, void* d_ws, size_t ws_size,
                              hipStream_t stream) {
  const float* x_c    = (const float*)d_in[0];
  const float* x_p    = (const float*)d_in[1];
  const int*   e_c2p  = (const int*)d_in[2];
  const int*   e_p2c  = (const int*)d_in[3];
  const float* W1l_r1 = (const float*)d_in[4];
  const float* b1_r1  = (const float*)d_in[5];
  const float* W1r_r1 = (const float*)d_in[6];
  const float* W1l_r2 = (const float*)d_in[7];
  const float* b1_r2  = (const float*)d_in[8];
  const float* W1r_r2 = (const float*)d_in[9];
  const float* W2l_r1 = (const float*)d_in[10];
  const float* b2_r1  = (const float*)d_in[11];
  const float* W2r_r1 = (const float*)d_in[12];
  // d_in[13..15] (W2l_r2, b2_r2, W2r_r2) unused: reference never builds h_c2
  const float* Wlin   = (const float*)d_in[16];
  const float* blin   = (const float*)d_in[17];

  const int NC = in_sizes[0] / 64;
  const int NP = in_sizes[1] / 64;
  const int E_ = in_sizes[2] / 2;

  // -------- workspace layout (floats) --------
  float* base   = (float*)d_ws;
  float* t_c    = base;                        // [NC,32] msg transform (c2p) -> reused as t2_c
  float* self_p = t_c    + (long)NC * 32;      // [NP,32] self (prod)        -> reused as self2_p
  float* t_p    = self_p + (long)NP * 32;      // [NP,32] msg transform (p2c) -> reused as agg2_p
  float* self_c = t_p    + (long)NP * 32;      // [NC,32] self (cust)
  float* agg_p  = self_c + (long)NC * 32;      // [NP,32] -> becomes h_p
  float* agg_c  = agg_p  + (long)NP * 32;      // [NC,32] -> becomes h_c
  float* cnt_p  = agg_c  + (long)NC * 32;      // [NP]
  float* cnt_c  = cnt_p  + NP;                 // [NC]   (agg_p..cnt_c contiguous)

  const int TB = 256;
  const int wavesPerBlock = TB / 32;

  // 1) zero agg_p, agg_c, cnt_p, cnt_c (contiguous)
  {
    long n = (long)NP * 32 + (long)NC * 32 + NP + NC;
    zero_kernel<<<(unsigned)((n + TB - 1) / TB), TB, 0, stream>>>(agg_p, n);
  }

  // 2) layer-1 node transforms (fused dual GEMM, fp32 WMMA)
  {
    int waves = NC / 16, blocks = (waves + wavesPerBlock - 1) / wavesPerBlock;
    gemm64_dual_wmma<<<blocks, TB, 0, stream>>>(x_c, W1l_r1, W1r_r2, b1_r2,
                                                t_c, self_c, NC);
  }
  {
    int waves = NP / 16, blocks = (waves + wavesPerBlock - 1) / wavesPerBlock;
    gemm64_dual_wmma<<<blocks, TB, 0, stream>>>(x_p, W1l_r2, W1r_r1, b1_r1,
                                                t_p, self_p, NP);
  }

  // 3) edge aggregation (mean numerators + counts), 8 lanes/edge
  {
    long nthr = (long)E_ * 8;
    unsigned blocks = (unsigned)((nthr + TB - 1) / TB);
    scatter_edges<<<blocks, TB, 0, stream>>>(e_c2p, E_, t_c, agg_p, cnt_p, 1);
    scatter_edges<<<blocks, TB, 0, stream>>>(e_p2c, E_, t_p, agg_c, cnt_c, 1);
  }

  // 4) h_p = relu(agg_p/cnt + self_p), h_c = relu(agg_c/cnt + self_c)  (in place)
  {
    long n = (long)NP * 32;
    finalize_kernel<<<(unsigned)((n + TB - 1) / TB), TB, 0, stream>>>(agg_p, self_p, cnt_p, n, 1);
    n = (long)NC * 32;
    finalize_kernel<<<(unsigned)((n + TB - 1) / TB), TB, 0, stream>>>(agg_c, self_c, cnt_c, n, 1);
  }

  // 5) layer-2 transforms: t2_c = h_c @ W2l_r1 ; self2_p = h_p @ W2r_r1 + b2_r1
  {
    int waves = NC / 16, blocks = (waves + wavesPerBlock - 1) / wavesPerBlock;
    gemm32_wmma<<<blocks, TB, 0, stream>>>(agg_c, W2l_r1, b2_r1, t_c, NC, 0);
  }
  {
    int waves = NP / 16, blocks = (waves + wavesPerBlock - 1) / wavesPerBlock;
    gemm32_wmma<<<blocks, TB, 0, stream>>>(agg_p, W2r_r1, b2_r1, self_p, NP, 1);
  }

  // 6) zero agg2_p (reuses t_p slot) and aggregate c2p again (counts reused)
  {
    long n = (long)NP * 32;
    zero_kernel<<<(unsigned)((n + TB - 1) / TB), TB, 0, stream>>>(t_p, n);
    long nthr = (long)E_ * 8;
    scatter_edges<<<(unsigned)((nthr + TB - 1) / TB), TB, 0, stream>>>(e_c2p, E_, t_c, t_p, cnt_p, 0);
  }

  // 7) head: mean + self + linear(32->1) + sigmoid
  {
    long nthr = (long)NP * 32;
    head_kernel<<<(unsigned)((nthr + TB - 1) / TB), TB, 0, stream>>>(
        t_p, self_p, cnt_p, Wlin, blin, (float*)d_out, NP);
  }
}